// SparseSoftmax_66718021976410
// MI455X (gfx1250) — compile-verified
//
#include <hip/hip_runtime.h>
#include <cstdint>

// ---------------- problem constants (reference shapes: X[4096, 8192]) -------
constexpr int kD      = 8192;          // row length
constexpr int kTPB    = 256;           // threads per block (8 waves, wave32)
constexpr int kPer    = kD / kTPB;     // 32 floats per thread, held in VGPRs
constexpr int kChunks = kPer / 4;      // 8 x b128 async loads per thread
constexpr int kIter   = 24;            // bisection steps

typedef __attribute__((ext_vector_type(2))) float v2f;
typedef __attribute__((ext_vector_type(8))) float v8f;

// ---------------- wave(32)-wide sum via V_WMMA_F32_16X16X4_F32 --------------
// A: lane value in VGPR0, 0 in VGPR1 ; B: all ones ; C = 0
// D[m,n] = sum_k A[m,k]  ->  sum of 8 D VGPRs + lane^16 add = full 32-lane sum
__device__ __forceinline__ float wave_sum32(float x) {
  v2f a;    a[0] = x;    a[1] = 0.0f;
  v2f ones; ones[0] = 1.0f; ones[1] = 1.0f;
  v8f c = {};
  v8f d = __builtin_amdgcn_wmma_f32_16x16x4_f32(
      /*neg_a=*/false, a, /*neg_b=*/false, ones,
      /*c_mod=*/(short)0, c, /*reuse_a=*/false, /*reuse_b=*/false);
  float t = ((d[0] + d[1]) + (d[2] + d[3])) + ((d[4] + d[5]) + (d[6] + d[7]));
  t += __shfl_xor(t, 16, 32);
  return t;  // identical in all 32 lanes
}

// ---------------- block-wide reductions (deterministic, fixed order) --------
__device__ __forceinline__ float block_sum(float v, float* red, int tid) {
  float w = wave_sum32(v);
  if ((tid & 31) == 0) red[tid >> 5] = w;
  __syncthreads();
  if (tid == 0) {
    float t = red[0];
#pragma unroll
    for (int i = 1; i < kTPB / 32; ++i) t += red[i];
    red[0] = t;
  }
  __syncthreads();
  float r = red[0];
  __syncthreads();  // protect red[] before next reduction reuses it
  return r;
}

__device__ __forceinline__ float block_max(float v, float* red, int tid) {
#pragma unroll
  for (int off = 16; off > 0; off >>= 1)
    v = fmaxf(v, __shfl_xor(v, off, 32));
  if ((tid & 31) == 0) red[tid >> 5] = v;
  __syncthreads();
  if (tid == 0) {
    float t = red[0];
#pragma unroll
    for (int i = 1; i < kTPB / 32; ++i) t = fmaxf(t, red[i]);
    red[0] = t;
  }
  __syncthreads();
  float r = red[0];
  __syncthreads();
  return r;
}

// ---------------- _p(z) = clip(z,0)^(1/(alpha-1)) ---------------------------
template <bool SQ>
__device__ __forceinline__ float pfun(float z, float cexp) {
  if (SQ) return z * z;  // alpha == 1.5 -> exponent exactly 2
  // hardware trans ops: v_log_f32 (log2) + v_exp_f32 (exp2)
  return (z > 0.0f) ? __builtin_amdgcn_exp2f(cexp * __builtin_amdgcn_logf(z)) : 0.0f;
}

// ---------------- per-row bisection solver (xa resident in VGPRs) -----------
template <bool SQ>
__device__ __forceinline__ void solve_row(const float (&xa)[kPer], float a, float am1,
                                          float cexp, float maxv, int tid, float* red,
                                          float& omega, float& dotp) {
  float tau_lo = maxv - 1.0f;                              // _gp(1, alpha)
  float tau_hi = maxv - powf(1.0f / (float)kD, am1);       // _gp(1/d, alpha)

  float f_lo;
  {
    float part = 0.0f;
#pragma unroll
    for (int j = 0; j < kPer; ++j) {
      float z = fmaxf(xa[j] - tau_lo, 0.0f);
      part += pfun<SQ>(z, cexp);
    }
    f_lo = block_sum(part, red, tid) - 1.0f;               // fixed, as in entmax
  }

  float dm = tau_hi - tau_lo;
  float tau_m = tau_lo;
#pragma unroll 1
  for (int it = 0; it < kIter; ++it) {
    dm *= 0.5f;
    tau_m = tau_lo + dm;
    float part = 0.0f;
#pragma unroll
    for (int j = 0; j < kPer; ++j) {
      float z = fmaxf(xa[j] - tau_m, 0.0f);
      part += pfun<SQ>(z, cexp);
    }
    float f_m = block_sum(part, red, tid) - 1.0f;
    if (f_m * f_lo >= 0.0f) tau_lo = tau_m;                // uniform branch
  }

  // epilogue: S1 = sum(p_m), Sdot = sum(p_m * xa), Sa = sum(p_m^a) = sum(p_m*z)
  float s1 = 0.0f, sdot = 0.0f, sa = 0.0f;
#pragma unroll
  for (int j = 0; j < kPer; ++j) {
    float z  = fmaxf(xa[j] - tau_m, 0.0f);
    float pm = pfun<SQ>(z, cexp);
    s1 += pm;
    sdot = fmaf(pm, xa[j], sdot);
    sa   = fmaf(pm, z, sa);
  }
  s1   = block_sum(s1,   red, tid);
  sdot = block_sum(sdot, red, tid);
  sa   = block_sum(sa,   red, tid);

  float s1a = SQ ? (s1 * sqrtf(s1)) : powf(s1, a);         // S1^alpha
  omega = (1.0f - sa / s1a) / (a * am1);                   // Tsallis term
  dotp  = sdot / (s1 * am1);                               // <p, X> = <p, xa>/(a-1)
}

// ---------------- kernel 1: one block per row -------------------------------
__global__ void __launch_bounds__(kTPB)
entmax_loss_rows(const float* __restrict__ X, const float* __restrict__ alpha,
                 const int* __restrict__ target, float* __restrict__ row_loss) {
  __shared__ float stage[kD];          // 32 KB staging (async global -> LDS)
  __shared__ float red[kTPB / 32];

  const int row = blockIdx.x;
  const int tid = threadIdx.x;
  const float a    = alpha[0];
  const float am1  = a - 1.0f;
  const float cexp = 1.0f / am1;
  const float* __restrict__ gbase = X + (size_t)row * kD;

  // CDNA5 async global->LDS staging: 8 x b128 per thread, coalesced chunks.
  // Each lane later reads back exactly the LDS bytes it requested, so
  // s_wait_asynccnt 0 (wave-local) is sufficient; no barrier needed.
#pragma unroll
  for (int c = 0; c < kChunks; ++c) {
    const int i4 = c * kTPB + tid;                       // float4 index in row
    uint32_t laddr = (uint32_t)(uintptr_t)(&stage[i4 * 4]);  // flat LDS: low 32b = offset
    unsigned long long gaddr = (unsigned long long)(uintptr_t)(gbase + i4 * 4);
    asm volatile("global_load_async_to_lds_b128 %0, %1, off"
                 :: "v"(laddr), "v"(gaddr) : "memory");
  }
  asm volatile("s_wait_asynccnt 0" ::: "memory");

  // readback into registers, scale by (alpha-1), track max
  float xa[kPer];
  float mx = -3.402823466e+38f;
#pragma unroll
  for (int c = 0; c < kChunks; ++c) {
    const int i4 = c * kTPB + tid;
    const float4 v = *reinterpret_cast<const float4*>(&stage[i4 * 4]);
    xa[c * 4 + 0] = v.x * am1;
    xa[c * 4 + 1] = v.y * am1;
    xa[c * 4 + 2] = v.z * am1;
    xa[c * 4 + 3] = v.w * am1;
    mx = fmaxf(mx, fmaxf(fmaxf(xa[c*4+0], xa[c*4+1]), fmaxf(xa[c*4+2], xa[c*4+3])));
  }
  const float maxv = block_max(mx, red, tid);

  float omega, dotp;
  if (cexp == 2.0f) solve_row<true >(xa, a, am1, cexp, maxv, tid, red, omega, dotp);
  else              solve_row<false>(xa, a, am1, cexp, maxv, tid, red, omega, dotp);

  if (tid == 0) {
    const int tgt = target[row];
    const float xt = gbase[tgt];
    row_loss[row] = omega + dotp - xt;
  }
}

// ---------------- kernel 2: deterministic final reduction -------------------
__global__ void __launch_bounds__(kTPB)
reduce_rows(const float* __restrict__ row_loss, float* __restrict__ out, int n) {
  __shared__ float red[kTPB / 32];
  float s = 0.0f;
  for (int i = threadIdx.x; i < n; i += kTPB) s += row_loss[i];
  float tot = block_sum(s, red, (int)threadIdx.x);
  if (threadIdx.x == 0) out[0] = tot;
}

// ---------------- launch ----------------------------------------------------
extern "C" void kernel_launch(void* const* d_in, const int* in_sizes, int n_in,
                              void* d_out, int out_size, void* d_ws, size_t ws_size,
                              hipStream_t stream) {
  const float* X      = (const float*)d_in[0];
  const float* alpha  = (const float*)d_in[1];
  const int*   target = (const int*)d_in[2];
  const int N = in_sizes[2];                  // 4096 rows; D fixed at 8192
  float* row_loss = (float*)d_ws;             // N floats of scratch

  entmax_loss_rows<<<dim3((unsigned)N), dim3(kTPB), 0, stream>>>(X, alpha, target, row_loss);
  reduce_rows<<<dim3(1), dim3(kTPB), 0, stream>>>(row_loss, (float*)d_out, N);
}